// GlobalModel_11338713662114
// MI455X (gfx1250) — compile-verified
//
#include <hip/hip_runtime.h>
#include <math.h>

#define TPB 256
typedef __attribute__((ext_vector_type(2))) float v2f;
typedef __attribute__((ext_vector_type(8))) float v8f;

static inline int nblk(long long t) { return (int)((t + TPB - 1) / TPB); }

// ---------------- fills ----------------
__global__ void fill_f32(float* p, float v, int n) {
  int i = blockIdx.x * TPB + threadIdx.x; if (i < n) p[i] = v;
}
__global__ void fill_u32(unsigned* p, unsigned v, int n) {
  int i = blockIdx.x * TPB + threadIdx.x; if (i < n) p[i] = v;
}

// ---------------- dense GEMMs ----------------
// stem: [n,3] @ [3,64]
__global__ void gemm3_64(const float* __restrict__ x, const float* __restrict__ W,
                         float* __restrict__ y, int n) {
  int tid = blockIdx.x * TPB + threadIdx.x;
  if (tid >= n * 64) return;
  int v = tid >> 6, c = tid & 63;
  const float* xr = x + (size_t)v * 3;
  y[tid] = xr[0] * W[c] + xr[1] * W[64 + c] + xr[2] * W[128 + c];
}

// main: [n,64] @ [64,64] via V_WMMA_F32_16X16X4_F32, W staged in LDS.
// 4 waves / block, each wave does a 16x64 strip (4 C tiles, K loop of 16 WMMAs).
__global__ void __launch_bounds__(128)
gemm64_wmma(const float* __restrict__ X, const float* __restrict__ W,
            float* __restrict__ Y, int n) {
  __shared__ float Ws[64 * 64];
  int t = threadIdx.x;
  for (int i = t; i < 64 * 64; i += 128) Ws[i] = W[i];
  __syncthreads();

  int wave = t >> 5;
  int lane = t & 31;
  int row0 = (blockIdx.x * 4 + wave) * 16;
  if (row0 >= n) return;            // n is a multiple of 16: whole-wave guard
  int lhalf = lane >> 4;            // 0: K lo pair / M,N 0..15 ; 1: K hi pair / M+8
  int lmod  = lane & 15;

  v8f acc[4] = {};
  const float* Xr = X + (size_t)(row0 + lmod) * 64;
  for (int k0 = 0; k0 < 64; k0 += 4) {
    v2f a;
    a.x = Xr[k0 + lhalf * 2 + 0];   // A 16x4 layout: lanes0-15 K=k0,k0+1; lanes16-31 K=k0+2,k0+3
    a.y = Xr[k0 + lhalf * 2 + 1];
#pragma unroll
    for (int nt = 0; nt < 4; ++nt) {
      v2f b;
      b.x = Ws[(k0 + lhalf * 2 + 0) * 64 + nt * 16 + lmod];
      b.y = Ws[(k0 + lhalf * 2 + 1) * 64 + nt * 16 + lmod];
      acc[nt] = __builtin_amdgcn_wmma_f32_16x16x4_f32(
          false, a, false, b, (short)0, acc[nt], false, false);
    }
  }
  // C/D layout: VGPR r -> (M = r + 8*lhalf, N = lmod)
#pragma unroll
  for (int nt = 0; nt < 4; ++nt) {
    float* Yp = Y + (size_t)row0 * 64 + nt * 16 + lmod;
#pragma unroll
    for (int r = 0; r < 8; ++r)
      Yp[(size_t)(r + lhalf * 8) * 64] = acc[nt][r];
  }
}

// head: [n,64] @ [64,1]
__global__ void dot64(const float* __restrict__ x, const float* __restrict__ w,
                      float* __restrict__ y, int n) {
  int v = blockIdx.x * TPB + threadIdx.x;
  if (v >= n) return;
  float s = 0.f;
#pragma unroll
  for (int c = 0; c < 64; ++c) s += x[(size_t)v * 64 + c] * w[c];
  y[v] = s;
}

// ---------------- GCN sparse part ----------------
__global__ void deg_k(const int* __restrict__ dst, const float* __restrict__ ew,
                      float* __restrict__ deg, int E_) {
  int e = blockIdx.x * TPB + threadIdx.x;
  if (e >= E_) return;
  atomicAdd(&deg[dst[e]], ew[e]);
}
__global__ void dinv_k(const float* __restrict__ deg, float* __restrict__ dinv,
                       float fill, int n) {
  int v = blockIdx.x * TPB + threadIdx.x;
  if (v >= n) return;
  dinv[v] = rsqrtf(deg[v] + fill);
}
__global__ void edge_norm(const int* __restrict__ src, const int* __restrict__ dst,
                          const float* __restrict__ ew, const float* __restrict__ dinv,
                          float* __restrict__ nrm, int E_) {
  int e = blockIdx.x * TPB + threadIdx.x;
  if (e >= E_) return;
  nrm[e] = dinv[src[e]] * ew[e] * dinv[dst[e]];
}
__global__ void agg_init64(const float* __restrict__ h, const float* __restrict__ dinv,
                           float fill, float* __restrict__ agg, int n) {
  int tid = blockIdx.x * TPB + threadIdx.x;
  if (tid >= n * 64) return;
  float di = dinv[tid >> 6];
  agg[tid] = fill * di * di * h[tid];
}
__global__ void agg_edges64(const float* __restrict__ h, const int* __restrict__ src,
                            const int* __restrict__ dst, const float* __restrict__ nrm,
                            float* __restrict__ agg, int E_) {
  int tid = blockIdx.x * TPB + threadIdx.x;
  int e = tid >> 6;
  if (e >= E_) return;
  int c = tid & 63;
  atomicAdd(&agg[(size_t)dst[e] * 64 + c], nrm[e] * h[(size_t)src[e] * 64 + c]);
}
__global__ void agg_init1(const float* __restrict__ h, const float* __restrict__ dinv,
                          float fill, float* __restrict__ agg, int n) {
  int v = blockIdx.x * TPB + threadIdx.x;
  if (v >= n) return;
  float di = dinv[v];
  agg[v] = fill * di * di * h[v];
}
__global__ void agg_edges1(const float* __restrict__ h, const int* __restrict__ src,
                           const int* __restrict__ dst, const float* __restrict__ nrm,
                           float* __restrict__ agg, int E_) {
  int e = blockIdx.x * TPB + threadIdx.x;
  if (e >= E_) return;
  atomicAdd(&agg[dst[e]], nrm[e] * h[src[e]]);
}

// ---------------- BatchNorm (training stats) + ReLU ----------------
__global__ void bn_reduce(const float* __restrict__ x, float* __restrict__ sums, int n) {
  int c = threadIdx.x & 63;
  int r0 = blockIdx.x * (TPB / 64) + (threadIdx.x >> 6);
  int rs = gridDim.x * (TPB / 64);
  float s = 0.f, s2 = 0.f;
  for (int r = r0; r < n; r += rs) {
    float v = x[(size_t)r * 64 + c];
    s += v; s2 += v * v;
  }
  atomicAdd(&sums[c], s);
  atomicAdd(&sums[64 + c], s2);
}
__global__ void bn_relu_k(const float* __restrict__ x, const float* __restrict__ sums,
                          const float* __restrict__ gamma, const float* __restrict__ beta,
                          float* __restrict__ y, int n) {
  int tid = blockIdx.x * TPB + threadIdx.x;
  if (tid >= n * 64) return;
  int c = tid & 63;
  float mean = sums[c] / (float)n;
  float var = sums[64 + c] / (float)n - mean * mean;
  float v = gamma[c] * (x[tid] - mean) * rsqrtf(var + 1e-5f) + beta[c];
  y[tid] = v > 0.f ? v : 0.f;
}

// ---------------- TopK pooling ----------------
__global__ void wnorm_k(const float* __restrict__ w, float* __restrict__ nrm) {
  if (threadIdx.x == 0 && blockIdx.x == 0) {
    float s = 0.f;
    for (int i = 0; i < 64; ++i) s += w[i] * w[i];
    nrm[0] = sqrtf(s);
  }
}
__global__ void scores_k(const float* __restrict__ x, const float* __restrict__ w,
                         const float* __restrict__ nrm, float* __restrict__ sc, int n) {
  int v = blockIdx.x * TPB + threadIdx.x;
  if (v >= n) return;
  float s = 0.f;
#pragma unroll
  for (int c = 0; c < 64; ++c) s += x[(size_t)v * 64 + c] * w[c];
  s /= nrm[0];
  sc[v] = s > 0.f ? s : 0.f;   // relu => scores >= 0 => uint-bit order == float order
}
__global__ void init_state(unsigned* st, unsigned k) {
  if (threadIdx.x == 0) { st[0] = 0u; st[1] = 0u; st[2] = k; }
}
__global__ void hist_k(const float* __restrict__ sc, const unsigned* __restrict__ st,
                       unsigned* __restrict__ hist, int n, int shift) {
  int i = blockIdx.x * TPB + threadIdx.x;
  if (i >= n) return;
  unsigned b = __float_as_uint(sc[i]);
  if ((b & st[1]) == st[0]) atomicAdd(&hist[(b >> shift) & 255u], 1u);
}
__global__ void pick_k(unsigned* st, const unsigned* __restrict__ hist, int shift) {
  if (threadIdx.x != 0) return;
  unsigned krem = st[2];
  unsigned cum = 0; int bin = 0;
  for (int b = 255; b >= 0; --b) {
    unsigned h = hist[b];
    if (cum + h >= krem) { bin = b; break; }
    cum += h;
  }
  st[0] |= ((unsigned)bin) << shift;
  st[1] |= 255u << shift;
  st[2] = krem - cum;
}
__global__ void cnt_k(const float* __restrict__ sc, const unsigned* __restrict__ st,
                      int mode, unsigned* __restrict__ bcnt, int n) {
  __shared__ unsigned lcnt;
  int i = blockIdx.x * TPB + threadIdx.x;
  if (threadIdx.x == 0) lcnt = 0u;
  __syncthreads();
  if (i < n) {
    unsigned b = __float_as_uint(sc[i]);
    bool f = (mode == 0) ? (b > st[0]) : (b == st[0]);
    if (f) atomicAdd(&lcnt, 1u);
  }
  __syncthreads();
  if (threadIdx.x == 0) bcnt[blockIdx.x] = lcnt;
}
__global__ void scan_k(const unsigned* __restrict__ bcnt, unsigned* __restrict__ bbase,
                       int nb, const unsigned* __restrict__ st, int mode, int kk) {
  if (threadIdx.x != 0 || blockIdx.x != 0) return;
  unsigned run = (mode == 0) ? 0u : ((unsigned)kk - st[2]);  // equal pass starts after strict-greaters
  for (int i = 0; i < nb; ++i) { bbase[i] = run; run += bcnt[i]; }
}
__global__ void scat_k(const float* __restrict__ sc, const unsigned* __restrict__ st,
                       int mode, const unsigned* __restrict__ bbase,
                       int* __restrict__ perm, int n, int kk) {
  __shared__ unsigned pref[TPB];
  int i = blockIdx.x * TPB + threadIdx.x;
  unsigned f = 0u;
  if (i < n) {
    unsigned b = __float_as_uint(sc[i]);
    f = (mode == 0) ? (unsigned)(b > st[0]) : (unsigned)(b == st[0]);
  }
  pref[threadIdx.x] = f;
  __syncthreads();
  for (int off = 1; off < TPB; off <<= 1) {
    unsigned add = (threadIdx.x >= (unsigned)off) ? pref[threadIdx.x - off] : 0u;
    __syncthreads();
    pref[threadIdx.x] += add;
    __syncthreads();
  }
  if (f) {
    unsigned pos = bbase[blockIdx.x] + pref[threadIdx.x] - 1u;
    if (pos < (unsigned)kk) perm[pos] = i;
  }
}
__global__ void pool_gather(const float* __restrict__ x, const int* __restrict__ perm,
                            const float* __restrict__ sc, float* __restrict__ xk, int k) {
  int tid = blockIdx.x * TPB + threadIdx.x;
  if (tid >= k * 64) return;
  int i = tid >> 6, c = tid & 63;
  int p = perm[i];
  xk[tid] = x[(size_t)p * 64 + c] * sc[p];
}
__global__ void set_map(const int* __restrict__ perm, int* __restrict__ mapping, int k) {
  int i = blockIdx.x * TPB + threadIdx.x;
  if (i >= k) return;
  mapping[perm[i]] = i;
}
__global__ void remap_edges(const int* __restrict__ src, const int* __restrict__ dst,
                            const float* __restrict__ ew, const int* __restrict__ mapping,
                            int* __restrict__ nsrc, int* __restrict__ ndst,
                            float* __restrict__ new_ew, int E_) {
  int e = blockIdx.x * TPB + threadIdx.x;
  if (e >= E_) return;
  int s = mapping[src[e]], d = mapping[dst[e]];
  bool valid = (s >= 0) && (d >= 0);
  nsrc[e] = valid ? s : 0;
  ndst[e] = valid ? d : 0;
  new_ew[e] = valid ? ew[e] : 0.f;
}

// ---------------- unpool / head ----------------
__global__ void up_add(const float* __restrict__ res, const float* __restrict__ xs,
                       const int* __restrict__ mapping, float* __restrict__ y, int n) {
  int tid = blockIdx.x * TPB + threadIdx.x;
  if (tid >= n * 64) return;
  int v = tid >> 6, c = tid & 63;
  int m = mapping[v];
  y[tid] = res[tid] + (m >= 0 ? xs[(size_t)m * 64 + c] : 0.f);
}
__global__ void sigmoid_k(const float* __restrict__ a, float* __restrict__ out, int n) {
  int v = blockIdx.x * TPB + threadIdx.x;
  if (v >= n) return;
  out[v] = 1.f / (1.f + expf(-a[v]));
}

// ---------------- host orchestration ----------------
extern "C" void kernel_launch(void* const* d_in, const int* in_sizes, int n_in,
                              void* d_out, int out_size, void* d_ws, size_t ws_size,
                              hipStream_t stream) {
  (void)n_in; (void)out_size; (void)ws_size;
  const float* x_in = (const float*)d_in[0];
  const int* eidx = (const int*)d_in[1];
  const int N = in_sizes[0] / 3;
  const int EE = in_sizes[1] / 2;
  const int* src0 = eidx;
  const int* dst0 = eidx + EE;
  const float* W_d[4] = {(const float*)d_in[2], (const float*)d_in[5], (const float*)d_in[8], (const float*)d_in[11]};
  const float* g_d[4] = {(const float*)d_in[3], (const float*)d_in[6], (const float*)d_in[9], (const float*)d_in[12]};
  const float* b_d[4] = {(const float*)d_in[4], (const float*)d_in[7], (const float*)d_in[10], (const float*)d_in[13]};
  const float* pw[3]  = {(const float*)d_in[14], (const float*)d_in[15], (const float*)d_in[16]};
  const float* W_u[2] = {(const float*)d_in[17], (const float*)d_in[20]};
  const float* g_u[2] = {(const float*)d_in[18], (const float*)d_in[21]};
  const float* b_u[2] = {(const float*)d_in[19], (const float*)d_in[22]};
  const float* W_out  = (const float*)d_in[23];

  int ns[4];
  ns[0] = N;
  for (int i = 0; i < 3; ++i) ns[i + 1] = (ns[i] + 1) / 2;

  // ---- bump allocator over d_ws (~180 MB total) ----
  char* wsp = (char*)d_ws; size_t off = 0;
  auto alloc = [&](size_t bytes) -> void* {
    void* p = wsp + off;
    off += (bytes + 255) & ~(size_t)255;
    return p;
  };
  float* xlev[4];
  for (int i = 0; i < 4; ++i) xlev[i] = (float*)alloc((size_t)ns[i] * 64 * 4);
  float* hbuf   = (float*)alloc((size_t)ns[0] * 64 * 4);
  float* aggbuf = (float*)alloc((size_t)ns[0] * 64 * 4);
  float* wA     = (float*)alloc((size_t)ns[0] * 64 * 4);
  float* wB     = (float*)alloc((size_t)ns[1] * 64 * 4);
  float* ew0     = (float*)alloc((size_t)EE * 4);
  float* normbuf = (float*)alloc((size_t)EE * 4);
  const int* gsrc[4]; const int* gdst[4]; const float* gew[4];
  int* msrc[4]; int* mdst[4]; float* mew[4];
  gsrc[0] = src0; gdst[0] = dst0; gew[0] = ew0;
  for (int i = 1; i < 4; ++i) {
    msrc[i] = (int*)alloc((size_t)EE * 4);
    mdst[i] = (int*)alloc((size_t)EE * 4);
    mew[i]  = (float*)alloc((size_t)EE * 4);
    gsrc[i] = msrc[i]; gdst[i] = mdst[i]; gew[i] = mew[i];
  }
  float* deg   = (float*)alloc((size_t)N * 4);
  float* dinv  = (float*)alloc((size_t)N * 4);
  float* scbuf = (float*)alloc((size_t)N * 4);
  float* h1buf = (float*)alloc((size_t)N * 4);
  int* map[3]; int* perm[3];
  for (int i = 0; i < 3; ++i) map[i]  = (int*)alloc((size_t)ns[i] * 4);
  for (int i = 0; i < 3; ++i) perm[i] = (int*)alloc((size_t)ns[i + 1] * 4);
  float* sums    = (float*)alloc(128 * 4);
  unsigned* hist = (unsigned*)alloc(256 * 4);
  unsigned* st   = (unsigned*)alloc(4 * 4);
  unsigned* bcnt  = (unsigned*)alloc(4096);
  unsigned* bbase = (unsigned*)alloc(4096);
  float* nrmbuf = (float*)alloc(4);

  auto conv64 = [&](const float* h, const int* s, const int* d, const float* ewp,
                    int n, float fillv) {
    fill_f32<<<nblk(n), TPB, 0, stream>>>(deg, 0.f, n);
    deg_k<<<nblk(EE), TPB, 0, stream>>>(d, ewp, deg, EE);
    dinv_k<<<nblk(n), TPB, 0, stream>>>(deg, dinv, fillv, n);
    edge_norm<<<nblk(EE), TPB, 0, stream>>>(s, d, ewp, dinv, normbuf, EE);
    agg_init64<<<nblk((long long)n * 64), TPB, 0, stream>>>(h, dinv, fillv, aggbuf, n);
    agg_edges64<<<nblk((long long)EE * 64), TPB, 0, stream>>>(h, s, d, normbuf, aggbuf, EE);
  };
  auto bnrelu = [&](const float* gamma, const float* beta, float* out, int n) {
    fill_f32<<<1, TPB, 0, stream>>>(sums, 0.f, 128);
    bn_reduce<<<256, TPB, 0, stream>>>(aggbuf, sums, n);
    bn_relu_k<<<nblk((long long)n * 64), TPB, 0, stream>>>(aggbuf, sums, gamma, beta, out, n);
  };
  auto topk = [&](const float* x, int n, int k, const float* pwp, int* permp, int* mapp,
                  const int* si, const int* di, const float* ei,
                  int* so, int* dso, float* eo, float* xk) {
    wnorm_k<<<1, 1, 0, stream>>>(pwp, nrmbuf);
    scores_k<<<nblk(n), TPB, 0, stream>>>(x, pwp, nrmbuf, scbuf, n);
    init_state<<<1, 1, 0, stream>>>(st, (unsigned)k);
    for (int shift = 24; shift >= 0; shift -= 8) {
      fill_u32<<<1, TPB, 0, stream>>>(hist, 0u, 256);
      hist_k<<<nblk(n), TPB, 0, stream>>>(scbuf, st, hist, n, shift);
      pick_k<<<1, 1, 0, stream>>>(st, hist, shift);
    }
    fill_u32<<<nblk(k), TPB, 0, stream>>>((unsigned*)permp, 0u, k);
    int nb = nblk(n);
    for (int mode = 0; mode < 2; ++mode) {
      cnt_k<<<nb, TPB, 0, stream>>>(scbuf, st, mode, bcnt, n);
      scan_k<<<1, 1, 0, stream>>>(bcnt, bbase, nb, st, mode, k);
      scat_k<<<nb, TPB, 0, stream>>>(scbuf, st, mode, bbase, permp, n, k);
    }
    pool_gather<<<nblk((long long)k * 64), TPB, 0, stream>>>(x, permp, scbuf, xk, k);
    fill_u32<<<nblk(n), TPB, 0, stream>>>((unsigned*)mapp, 0xFFFFFFFFu, n);
    set_map<<<nblk(k), TPB, 0, stream>>>(permp, mapp, k);
    remap_edges<<<nblk(EE), TPB, 0, stream>>>(si, di, ei, mapp, so, dso, eo, EE);
  };

  // ---------- down path ----------
  fill_f32<<<nblk(EE), TPB, 0, stream>>>(ew0, 1.f, EE);
  gemm3_64<<<nblk((long long)N * 64), TPB, 0, stream>>>(x_in, W_d[0], hbuf, N);
  conv64(hbuf, gsrc[0], gdst[0], gew[0], ns[0], 2.f);
  bnrelu(g_d[0], b_d[0], xlev[0], ns[0]);

  for (int i = 0; i < 3; ++i) {
    topk(xlev[i], ns[i], ns[i + 1], pw[i], perm[i], map[i],
         gsrc[i], gdst[i], gew[i], msrc[i + 1], mdst[i + 1], mew[i + 1], wA);
    gemm64_wmma<<<(ns[i + 1] + 63) / 64, 128, 0, stream>>>(wA, W_d[i + 1], hbuf, ns[i + 1]);
    conv64(hbuf, gsrc[i + 1], gdst[i + 1], gew[i + 1], ns[i + 1], 2.f);
    bnrelu(g_d[i + 1], b_d[i + 1], xlev[i + 1], ns[i + 1]);
  }

  // ---------- up path ----------
  float* cur = xlev[3];
  for (int i = 0; i < 2; ++i) {
    int j = 2 - i;  // residual level: x2 then x1
    up_add<<<nblk((long long)ns[j] * 64), TPB, 0, stream>>>(xlev[j], cur, map[j], wA, ns[j]);
    gemm64_wmma<<<(ns[j] + 63) / 64, 128, 0, stream>>>(wA, W_u[i], hbuf, ns[j]);
    conv64(hbuf, gsrc[j], gdst[j], gew[j], ns[j], 2.f);
    bnrelu(g_u[i], b_u[i], wB, ns[j]);
    cur = wB;
  }

  // ---------- final level: unpool + gcn_conv(W_out, fill=1) + sigmoid ----------
  up_add<<<nblk((long long)ns[0] * 64), TPB, 0, stream>>>(xlev[0], cur, map[0], wA, ns[0]);
  dot64<<<nblk(ns[0]), TPB, 0, stream>>>(wA, W_out, h1buf, ns[0]);
  fill_f32<<<nblk(ns[0]), TPB, 0, stream>>>(deg, 0.f, ns[0]);
  deg_k<<<nblk(EE), TPB, 0, stream>>>(gdst[0], gew[0], deg, EE);
  dinv_k<<<nblk(ns[0]), TPB, 0, stream>>>(deg, dinv, 1.f, ns[0]);
  edge_norm<<<nblk(EE), TPB, 0, stream>>>(gsrc[0], gdst[0], gew[0], dinv, normbuf, EE);
  agg_init1<<<nblk(ns[0]), TPB, 0, stream>>>(h1buf, dinv, 1.f, scbuf, ns[0]);
  agg_edges1<<<nblk(EE), TPB, 0, stream>>>(h1buf, gsrc[0], gdst[0], normbuf, scbuf, EE);
  sigmoid_k<<<nblk(ns[0]), TPB, 0, stream>>>(scbuf, (float*)d_out, ns[0]);
}